// Net_15238543966692
// MI455X (gfx1250) — compile-verified
//
#include <hip/hip_runtime.h>

// ---------------------------------------------------------------------------
// PointNet++ front-end for MI455X (gfx1250, wave32):
//   1) fps_kernel   : farthest point sampling, 1 block (1024 thr) per batch,
//                     coords + running min-dist fully register resident.
//   2) qball_kernel : query_ball_point(radius=0.5, nsample=1) via
//                     V_WMMA_F32_16X16X4_F32 computing the FULL squared
//                     distance in one matrix op:
//                       A = [-2qx, -2qy, -2qz, 1]   (16x4)
//                       B = [ px,   py,   pz, |p|^2] (4x16)
//                       C = |q|^2 broadcast per row  (16x16)
//                       D = A*B + C = |q-p|^2
// ---------------------------------------------------------------------------

typedef float v2f __attribute__((ext_vector_type(2)));
typedef float v8f __attribute__((ext_vector_type(8)));

#define FPS_THREADS 1024
#define FPS_PPT     16          // points per thread: N / 1024 (N = 16384)

// Block-wide argmax with first-index tie-break (matches jnp.argmax).
// All 1024 threads must call. Returns winning index (uniform).
__device__ __forceinline__ int block_argmax(float lv, int ln,
                                            float* sVal, int* sIdx, int* sFar,
                                            int lane, int wave)
{
    // wave32 reduction
    #pragma unroll
    for (int m = 16; m >= 1; m >>= 1) {
        float ov = __shfl_xor(lv, m, 32);
        int   on = __shfl_xor(ln, m, 32);
        if (ov > lv || (ov == lv && on < ln)) { lv = ov; ln = on; }
    }
    if (lane == 0) { sVal[wave] = lv; sIdx[wave] = ln; }
    __syncthreads();
    if (wave == 0) {
        float v = sVal[lane]; int n = sIdx[lane];
        #pragma unroll
        for (int m = 16; m >= 1; m >>= 1) {
            float ov = __shfl_xor(v, m, 32);
            int   on = __shfl_xor(n, m, 32);
            if (ov > v || (ov == v && on < n)) { v = ov; n = on; }
        }
        if (lane == 0) *sFar = n;
    }
    __syncthreads();
    return *sFar;
}

__global__ __launch_bounds__(FPS_THREADS)
void fps_kernel(const float* __restrict__ xyz,   // [B,3,N]
                float* __restrict__ new_xyz,      // [B,S,3] (workspace)
                int N, int S)
{
    const int b = blockIdx.x;
    const int t = threadIdx.x;
    const int lane = t & 31;
    const int wave = t >> 5;

    const float* X = xyz + (size_t)b * 3 * N;
    const float* Y = X + N;
    const float* Z = Y + N;

    // register-resident point data + running min distance (stride-1024 layout)
    float px[FPS_PPT], py[FPS_PPT], pz[FPS_PPT], dist[FPS_PPT];
    #pragma unroll
    for (int i = 0; i < FPS_PPT; ++i) {
        const int n = t + (i << 10);
        px[i] = X[n]; py[i] = Y[n]; pz[i] = Z[n];
        dist[i] = 1e10f;                     // BIG
    }

    __shared__ float sVal[32];
    __shared__ int   sIdx[32];
    __shared__ int   sFar;

    // initial farthest = argmax over x coordinate
    float lv = -1e30f; int ln = 0;
    #pragma unroll
    for (int i = 0; i < FPS_PPT; ++i) {
        const int n = t + (i << 10);
        if (px[i] > lv) { lv = px[i]; ln = n; }
    }
    int far = block_argmax(lv, ln, sVal, sIdx, &sFar, lane, wave);

    for (int k = 0; k < S; ++k) {
        // record PRE-update farthest (coords broadcast via same-address load)
        const float bx = X[far], by = Y[far], bz = Z[far];
        if (t == 0) {
            float* o = new_xyz + ((size_t)b * S + k) * 3;
            o[0] = bx; o[1] = by; o[2] = bz;
        }
        if (k == S - 1) break;

        float mv = -1e30f; int mn = 0;
        #pragma unroll
        for (int i = 0; i < FPS_PPT; ++i) {
            const float dx = px[i] - bx;
            const float dy = py[i] - by;
            const float dz = pz[i] - bz;
            const float d  = dx * dx + dy * dy + dz * dz;
            const float nd = fminf(dist[i], d);
            dist[i] = nd;
            if (nd > mv) { mv = nd; mn = t + (i << 10); }
        }
        far = block_argmax(mv, mn, sVal, sIdx, &sFar, lane, wave);
    }
}

// One wave (32 threads) handles 16 query points against all N points.
// The WMMA produces the full squared distance: D = A*B + C (see header).
__global__ __launch_bounds__(32)
void qball_kernel(const float* __restrict__ xyz,      // [B,3,N]
                  const float* __restrict__ new_xyz,  // [B,S,3]
                  int* __restrict__ out,              // [B,S,1]
                  int N, int S)
{
    const int groups = S >> 4;                 // query groups per batch
    const int b    = blockIdx.x / groups;
    const int qg   = blockIdx.x % groups;
    const int lane = threadIdx.x;              // 0..31
    const int half = lane >> 4;                // 0: K=0,1  1: K=2,3
    const int col  = lane & 15;

    const float* X = xyz + (size_t)b * 3 * N;
    const float* Y = X + N;
    const float* Z = Y + N;

    // Query point for matrix row M = col
    const float* q = new_xyz + ((size_t)b * S + (size_t)qg * 16 + col) * 3;
    const float qx = q[0], qy = q[1], qz = q[2];
    const float qnM = qx * qx + qy * qy + qz * qz;

    // A (16x4 f32): lanes 0-15 hold K={0,1}=(-2x,-2y); 16-31 K={2,3}=(-2z, 1)
    v2f a;
    a.x = half ? (-2.0f * qz) : (-2.0f * qx);
    a.y = half ? 1.0f         : (-2.0f * qy);

    // C (16x16 f32) = |q|^2 broadcast along columns:
    // D/C element v -> row M = v + 8*half, held by lane (v + 8*half) of group 0-15.
    v8f c;
    #pragma unroll
    for (int v = 0; v < 8; ++v)
        c[v] = __shfl(qnM, v + (half << 3), 32);

    int best[8];
    #pragma unroll
    for (int v = 0; v < 8; ++v) best[v] = N;   // sentinel: "no point in ball"

    const float r2 = 0.25f;                    // RADIUS^2

    for (int tb = 0; tb < N; tb += 16) {
        const int n = tb + col;
        const float pxv = X[n], pyv = Y[n], pzv = Z[n];
        const float pn  = pxv * pxv + pyv * pyv + pzv * pzv;

        // B (4x16 f32): lanes 0-15 K={0,1}=(x,y); lanes 16-31 K={2,3}=(z,|p|^2)
        v2f bm;
        bm.x = half ? pzv : pxv;
        bm.y = half ? pn  : pyv;

        // D = |q - p|^2 for all 16x16 pairs, straight out of the matrix unit
        v8f d = __builtin_amdgcn_wmma_f32_16x16x4_f32(
            /*neg_a=*/false, a, /*neg_b=*/false, bm,
            /*c_mod=*/(short)0, c, /*reuse_a=*/false, /*reuse_b=*/false);

        #pragma unroll
        for (int v = 0; v < 8; ++v) {
            const int cand = (d[v] <= r2) ? n : N;
            best[v] = (cand < best[v]) ? cand : best[v];
        }
    }

    // Min-index across the 16 columns (stay within each 16-lane half)
    #pragma unroll
    for (int v = 0; v < 8; ++v) {
        #pragma unroll
        for (int m = 1; m < 16; m <<= 1) {
            const int o = __shfl_xor(best[v], m, 32);
            best[v] = (o < best[v]) ? o : best[v];
        }
    }

    // lane 0 writes rows 0..7, lane 16 writes rows 8..15
    if (col == 0) {
        int* o = out + (size_t)b * S + (size_t)qg * 16 + (half << 3);
        #pragma unroll
        for (int v = 0; v < 8; ++v) o[v] = best[v];
    }
}

extern "C" void kernel_launch(void* const* d_in, const int* in_sizes, int n_in,
                              void* d_out, int out_size, void* d_ws, size_t ws_size,
                              hipStream_t stream)
{
    const float* xyz = (const float*)d_in[0];          // [B,3,N] float32
    // d_in[1] (cls_label) unused; d_in[2] (npoint) fixed by output shape.

    const int B = in_sizes[1] / 16;                    // cls_label is [B,16]
    const int N = in_sizes[0] / (3 * B);               // 16384
    const int S = out_size / B;                        // npoint = 1024

    float* new_xyz = (float*)d_ws;                     // B*S*3 floats (48 KB)

    fps_kernel<<<B, FPS_THREADS, 0, stream>>>(xyz, new_xyz, N, S);

    const int qblocks = B * (S / 16);
    qball_kernel<<<qblocks, 32, 0, stream>>>(xyz, new_xyz, (int*)d_out, N, S);
}